// ReChroma_5549097747038
// MI455X (gfx1250) — compile-verified
//
#include <hip/hip_runtime.h>
#include <math.h>
#include <stdint.h>

typedef __attribute__((ext_vector_type(2))) float v2f;
typedef __attribute__((ext_vector_type(8))) float v8f;

#define LSTRIDE 132          // padded LDS row stride (floats): conflict-free frags
                             // 132*4 = 528 B = 33*16 -> vec4 chunks stay 16B aligned
#define NPIX    16384        // 128*128
#define NELEM   262144       // 16*128*128

// ---------------------------------------------------------------------------
// Build the 128x128 reflect-padded Gaussian conv operator G in ws.
// ---------------------------------------------------------------------------
__global__ __launch_bounds__(128) void build_gauss_mat(const int* sigma_p,
                                                       const int* ks_p,
                                                       float* __restrict__ G) {
    const int i = threadIdx.x;             // row 0..127
    int ks = ks_p[0];
    if (ks <= 0) ks = 13;
    if ((ks & 1) == 0) ks += 1;            // reference: even -> odd
    if (ks > 31) ks = 31;
    const float sigma = (float)sigma_p[0];
    const int pad = ks >> 1;

    float s = 0.f;
    for (int k = 0; k < ks; ++k) {
        float d = (float)(k - pad) / sigma;
        s += __expf(-0.5f * d * d);
    }
    const float inv = 1.f / s;

    float* row = G + i * 128;
    for (int j = 0; j < 128; ++j) row[j] = 0.f;
    for (int k = 0; k < ks; ++k) {
        float d = (float)(k - pad) / sigma;
        float w = __expf(-0.5f * d * d) * inv;
        int p = i + k - pad;
        if (p < 0)   p = -p;               // jnp.pad mode='reflect' (no edge dup)
        if (p > 127) p = 254 - p;
        row[p] += w;
    }
}

// ---------------------------------------------------------------------------
// One workgroup per (field, channel): D = G * X * G using
// v_wmma_f32_16x16x4_f32. fields layout: [field][ch][h][w], field in
// {0: mean(c), 1: E[c^2], 2: mean(s), 3: E[s^2]}.
// ---------------------------------------------------------------------------
__global__ __launch_bounds__(256) void adain_conv_wmma(
    const float* __restrict__ content, const float* __restrict__ style,
    const float* __restrict__ G, float* __restrict__ fields) {

    __shared__ float ldsX[128 * LSTRIDE];
    __shared__ float ldsT[128 * LSTRIDE];

    const int blk   = blockIdx.x;          // 0..63
    const int field = blk >> 4;            // 0..3
    const int ch    = blk & 15;
    const float* src = ((field < 2) ? content : style) + ch * NPIX;
    float* dst = fields + (size_t)blk * NPIX;
    const bool sq = (field & 1) != 0;      // block-uniform

    const int tid = threadIdx.x;

    // Phase A: stage X into LDS (padded stride).
    if (!sq) {
        // Raw field: async global->LDS DMA, 16B per lane-op, no VGPR round-trip.
        for (int i4 = tid; i4 < NPIX / 4; i4 += 256) {
            const int row = i4 >> 5;                 // 32 vec4 per row
            const int col = (i4 & 31) * 4;
            uint32_t lds_addr =
                (uint32_t)(uintptr_t)(&ldsX[row * LSTRIDE + col]);
            uint64_t gaddr = (uint64_t)(uintptr_t)(src + row * 128 + col);
            asm volatile("global_load_async_to_lds_b128 %0, %1, off"
                         :: "v"(lds_addr), "v"(gaddr)
                         : "memory");
        }
        asm volatile("s_wait_asynccnt 0x0" ::: "memory");
    } else {
        // Squared field: must square in VALU on the way in.
        for (int i = tid; i < NPIX; i += 256) {
            float v = src[i];
            ldsX[(i >> 7) * LSTRIDE + (i & 127)] = v * v;
        }
    }
    __syncthreads();

    const int lane = tid & 31;
    const int wave = tid >> 5;             // 8 waves per workgroup
    const int m    = lane & 15;            // M (A) / N (B,C,D) index
    const int hf   = lane >> 4;            // lane half select
    const int koff = hf * 2;               // K sub-offset per ISA A/B layout

    // Phase B: T = G * X   (A from global G, B from LDS X)
    for (int t = wave; t < 64; t += 8) {
        const int bi = (t >> 3) * 16;      // output row base
        const int bj = (t & 7) * 16;       // output col base
        v8f acc = {};
#pragma unroll 8
        for (int kk = 0; kk < 128; kk += 4) {
            v2f a, b;
            const float* gp = G + (bi + m) * 128 + kk + koff;
            a.x = gp[0];
            a.y = gp[1];
            b.x = ldsX[(kk + koff)     * LSTRIDE + bj + m];
            b.y = ldsX[(kk + koff + 1) * LSTRIDE + bj + m];
            acc = __builtin_amdgcn_wmma_f32_16x16x4_f32(
                false, a, false, b, (short)0, acc, false, false);
        }
#pragma unroll
        for (int v = 0; v < 8; ++v)
            ldsT[(bi + v + 8 * hf) * LSTRIDE + bj + m] = acc[v];
    }
    __syncthreads();

    // Phase C: D = T * G   (A from LDS T, B from global G)
    for (int t = wave; t < 64; t += 8) {
        const int bi = (t >> 3) * 16;
        const int bj = (t & 7) * 16;
        v8f acc = {};
#pragma unroll 8
        for (int kk = 0; kk < 128; kk += 4) {
            v2f a, b;
            const float* tp = ldsT + (bi + m) * LSTRIDE + kk + koff;
            a.x = tp[0];
            a.y = tp[1];
            b.x = G[(kk + koff)     * 128 + bj + m];
            b.y = G[(kk + koff + 1) * 128 + bj + m];
            acc = __builtin_amdgcn_wmma_f32_16x16x4_f32(
                false, a, false, b, (short)0, acc, false, false);
        }
#pragma unroll
        for (int v = 0; v < 8; ++v)
            dst[(bi + v + 8 * hf) * 128 + bj + m] = acc[v];
    }
}

// ---------------------------------------------------------------------------
// Elementwise AdaIN combine: (c - mc)/(sqrt(var_c)+eps) * (sqrt(var_s)+eps) + ms
// ---------------------------------------------------------------------------
__global__ __launch_bounds__(256) void adain_finalize(
    const float* __restrict__ content, const float* __restrict__ fields,
    float* __restrict__ out, int n) {
    int i = blockIdx.x * 256 + threadIdx.x;
    if (i >= n) return;
    const float eps = 1e-5f;
    float c   = content[i];
    float mc  = fields[i];
    float mc2 = fields[i + 1 * NELEM];
    float ms  = fields[i + 2 * NELEM];
    float ms2 = fields[i + 3 * NELEM];
    float cstd = sqrtf(fmaxf(mc2 - mc * mc, 0.f)) + eps;
    float sstd = sqrtf(fmaxf(ms2 - ms * ms, 0.f)) + eps;
    out[i] = (c - mc) / cstd * sstd + ms;
}

extern "C" void kernel_launch(void* const* d_in, const int* in_sizes, int n_in,
                              void* d_out, int out_size, void* d_ws, size_t ws_size,
                              hipStream_t stream) {
    const float* content = (const float*)d_in[0];
    const float* style   = (const float*)d_in[1];
    const int*   sigma   = (const int*)d_in[2];
    const int*   ksz     = (const int*)d_in[3];

    float* G      = (float*)d_ws;               // 128*128 floats = 64 KB
    float* fields = (float*)d_ws + 128 * 128;   // 4 * 16*128*128 floats = 4 MB

    build_gauss_mat<<<1, 128, 0, stream>>>(sigma, ksz, G);
    adain_conv_wmma<<<64, 256, 0, stream>>>(content, style, G, fields);
    adain_finalize<<<(NELEM + 255) / 256, 256, 0, stream>>>(content, fields,
                                                            (float*)d_out, NELEM);
}